// ImplicitGraph_52733608461008
// MI455X (gfx1250) — compile-verified
//
#include <hip/hip_runtime.h>
#include <hip/hip_bf16.h>
#include <stdint.h>

typedef float v2f __attribute__((ext_vector_type(2)));
typedef float v8f __attribute__((ext_vector_type(8)));

#define MDIM     128
#define KAPPA_V  0.99f
#define TOL_V    3e-6f
#define MAX_ITERS 300

// ---------------------------------------------------------------------------
// Row-wise L1-ball projection of W (128x128), writing W^T (Wt[k][m] = Wp[m][k])
// One block (128 threads) per row: bitonic sort desc + prefix sum + threshold.
// ---------------------------------------------------------------------------
__global__ void proj_w_kernel(const float* __restrict__ W, float* __restrict__ Wt) {
    __shared__ float s[128];
    __shared__ float cs[128];
    __shared__ int cnt;
    const int r = blockIdx.x;
    const int t = threadIdx.x;
    const float w  = W[r * 128 + t];
    const float aw = fabsf(w);
    s[t] = aw;
    if (t == 0) cnt = 0;
    __syncthreads();
    // bitonic sort, descending
    for (int k = 2; k <= 128; k <<= 1) {
        for (int j = k >> 1; j > 0; j >>= 1) {
            int ixj = t ^ j;
            if (ixj > t) {
                float a = s[t], b = s[ixj];
                bool descRegion = ((t & k) == 0);
                bool doSwap = descRegion ? (a < b) : (a > b);
                if (doSwap) { s[t] = b; s[ixj] = a; }
            }
            __syncthreads();
        }
    }
    // inclusive prefix sum of sorted values
    cs[t] = s[t];
    __syncthreads();
    for (int off = 1; off < 128; off <<= 1) {
        float v = (t >= off) ? cs[t - off] : 0.0f;
        __syncthreads();
        cs[t] += v;
        __syncthreads();
    }
    const float rowsum = cs[127];
    const float css = cs[t] - KAPPA_V;
    if (s[t] - css / (float)(t + 1) > 0.0f) atomicAdd(&cnt, 1);
    __syncthreads();
    const int rho = cnt;
    float res = w;
    if (rowsum > KAPPA_V) {
        int rr = (rho > 0) ? rho : 1;
        float theta = (cs[rr - 1] - KAPPA_V) / (float)rr;
        float p = fmaxf(aw - theta, 0.0f);
        res = (w >= 0.0f) ? p : -p;
    }
    Wt[t * 128 + r] = res;   // transposed store
}

// ---------------------------------------------------------------------------
// Tiled transpose: dst[c*R + r] = src[r*C + c]   (src is R x C, row-major)
// ---------------------------------------------------------------------------
__global__ void transpose_kernel(const float* __restrict__ src, float* __restrict__ dst,
                                 int R, int C) {
    __shared__ float tile[32][33];
    const int bx = blockIdx.x * 32;   // C dimension
    const int by = blockIdx.y * 32;   // R dimension
    const int tx = threadIdx.x, ty = threadIdx.y;  // 32 x 8
    for (int i = ty; i < 32; i += 8) {
        int r = by + i, c = bx + tx;
        tile[i][tx] = (r < R && c < C) ? src[(size_t)r * C + c] : 0.0f;
    }
    __syncthreads();
    for (int i = ty; i < 32; i += 8) {
        int c = bx + i, r = by + tx;
        if (c < C && r < R) dst[(size_t)c * R + r] = tile[tx][i];
    }
}

// ---------------------------------------------------------------------------
// Node-major GEMM via fp32 WMMA:  Yt[n][m] = sum_k Xt[n][k] * Wt[k][m]
// Block: 128 nodes x 128 outputs, 8 waves; wave w owns node band [16w,16w+16).
//
// LDS W layout is k-PAIRED: ldsW[(k>>1)*256 + m*2 + (k&1)] = Wt[k][m], so one
// WMMA B-fragment (rows ka, ka+1 at column mb+l) is 8 contiguous, 8B-aligned
// bytes -> single ds_load_b64 with a small immediate offset off a per-mt base.
//
// ISA layouts (7.12.2):
//   A 16x4: lane<16 -> row M=lane, k0/k0+1 ; lane>=16 -> k0+2/k0+3
//   B 4x16: lane<16 -> col N=lane, rows k0/k0+1 ; lane>=16 -> k0+2/k0+3
//   D     : VGPR j, lane<16 -> (M=j, N=lane) ; lane>=16 -> (M=j+8, N=lane-16)
// ---------------------------------------------------------------------------
__global__ __launch_bounds__(256)
void gemm_nodemajor_kernel(const float* __restrict__ Xt, const float* __restrict__ Wt,
                           float* __restrict__ Yt, int N, const int* __restrict__ flag) {
    if (flag && *flag) return;
    __shared__ __align__(16) float ldsW[128 * 128];   // k-paired layout
    __shared__ __align__(16) float ldsX[128 * 132];   // padded stride 132
    const int tid = threadIdx.x;
    const int nodeBase = blockIdx.x * 128;

    // stage W with k-pairing: ldsW[(k>>1)*256 + m*2 + (k&1)]
    for (int idx = tid; idx < 128 * 32; idx += 256) {
        int k = idx >> 5, q = idx & 31;
        float4 wv = ((const float4*)Wt)[idx];          // Wt[k][q*4 .. q*4+3]
        int base = (k >> 1) * 256 + (k & 1);
        ldsW[base + (q * 4 + 0) * 2] = wv.x;
        ldsW[base + (q * 4 + 1) * 2] = wv.y;
        ldsW[base + (q * 4 + 2) * 2] = wv.z;
        ldsW[base + (q * 4 + 3) * 2] = wv.w;
    }
    // stage X tile (zero-fill rows beyond N so WMMA runs with full EXEC)
    for (int idx = tid; idx < 128 * 32; idx += 256) {
        int row = idx >> 5, q = idx & 31;
        int node = nodeBase + row;
        float4 xv = {0.f, 0.f, 0.f, 0.f};
        if (node < N) xv = *(const float4*)(Xt + (size_t)node * 128 + q * 4);
        *(float4*)(ldsX + row * 132 + q * 4) = xv;
    }
    __syncthreads();

    const int wave = tid >> 5;
    const int lane = tid & 31;
    const int l  = lane & 15;
    const int lh = lane >> 4;
    const int nodeSub = wave * 16;
    const float* arow = &ldsX[(nodeSub + l) * 132 + 2 * lh];

    for (int mt = 0; mt < 8; ++mt) {
        const int mb = mt * 16;
        const float* brow = &ldsW[(mb + l) * 2];       // per-mt B base pointer
        v8f acc = {};
#pragma unroll
        for (int k0 = 0; k0 < 128; k0 += 4) {
            const int p = (k0 >> 1) + lh;              // pair index (ka = 2p)
            v2f a = *(const v2f*)(arow + k0);          // ds_load_b64
            v2f b = *(const v2f*)(brow + p * 256);     // ds_load_b64, imm offset
            acc = __builtin_amdgcn_wmma_f32_16x16x4_f32(
                false, a, false, b, (short)0, acc, false, false);
        }
        const int nodeLocal = nodeBase + nodeSub + 8 * lh;
#pragma unroll
        for (int j = 0; j < 8; ++j) {
            int node = nodeLocal + j;
            if (node < N) Yt[(size_t)node * 128 + mb + l] = acc[j];
        }
    }
}

// ---------------------------------------------------------------------------
// Edge-parallel SpMM scatter: Acc[c][:] += vals[e] * Yt[r][:]  (wave per edge)
// Contiguous 512B gather + 128 f32 atomics per edge (node-major layout).
// ---------------------------------------------------------------------------
__global__ void spmm_kernel(const int* __restrict__ rows, const int* __restrict__ cols,
                            const float* __restrict__ vals, const float* __restrict__ Yt,
                            float* __restrict__ Acc, int nnz, const int* __restrict__ flag) {
    if (flag && *flag) return;
    const int lane = threadIdx.x & 31;
    int wid = (blockIdx.x * blockDim.x + threadIdx.x) >> 5;
    const int nwaves = (gridDim.x * blockDim.x) >> 5;
    for (int e = wid; e < nnz; e += nwaves) {
        const int r = rows[e];
        const int c = cols[e];
        const float v = vals[e];
        float4 d = *(const float4*)(Yt + (size_t)r * 128 + lane * 4);
        float* dst = Acc + (size_t)c * 128 + lane * 4;
        atomicAdd(dst + 0, v * d.x);
        atomicAdd(dst + 1, v * d.y);
        atomicAdd(dst + 2, v * d.z);
        atomicAdd(dst + 3, v * d.w);
    }
}

// ---------------------------------------------------------------------------
__global__ void zero_kernel(float* __restrict__ buf, size_t n,
                            unsigned* errMax, const int* __restrict__ flag) {
    if (flag && *flag) return;
    size_t i = (size_t)blockIdx.x * blockDim.x + threadIdx.x;
    const size_t stride = (size_t)gridDim.x * blockDim.x;
    for (; i < n; i += stride) buf[i] = 0.0f;
    if (blockIdx.x == 0 && threadIdx.x == 0 && errMax) *errMax = 0u;
}

__global__ void update_kernel(float* __restrict__ Xt, const float* __restrict__ Acc,
                              const float* __restrict__ Bt, unsigned* errMax,
                              const int* __restrict__ flag, size_t n) {
    if (*flag) return;
    float local = 0.0f;
    size_t i = (size_t)blockIdx.x * blockDim.x + threadIdx.x;
    const size_t stride = (size_t)gridDim.x * blockDim.x;
    for (; i < n; i += stride) {
        float xo = Xt[i];
        float xn = fmaxf(Acc[i] + Bt[i], 0.0f);
        Xt[i] = xn;
        local = fmaxf(local, fabsf(xn - xo));
    }
    __shared__ float red[256];
    red[threadIdx.x] = local;
    __syncthreads();
    for (int o = 128; o > 0; o >>= 1) {
        if (threadIdx.x < o) red[threadIdx.x] = fmaxf(red[threadIdx.x], red[threadIdx.x + o]);
        __syncthreads();
    }
    if (threadIdx.x == 0) atomicMax(errMax, __float_as_uint(red[0]));
}

__global__ void check_kernel(const unsigned* __restrict__ errMax, int* __restrict__ flag) {
    if (*flag) return;
    if (__uint_as_float(*errMax) < TOL_V) *flag = 1;
}

__global__ void init_kernel(int* flag, unsigned* errMax) {
    *flag = 0;
    *errMax = 0u;
}

// ---------------------------------------------------------------------------
extern "C" void kernel_launch(void* const* d_in, const int* in_sizes, int n_in,
                              void* d_out, int out_size, void* d_ws, size_t ws_size,
                              hipStream_t stream) {
    const float* X0   = (const float*)d_in[0];
    const float* U    = (const float*)d_in[1];
    const int*   rows = (const int*)d_in[2];
    const int*   cols = (const int*)d_in[3];
    const float* vals = (const float*)d_in[4];
    const float* W    = (const float*)d_in[5];
    const float* Om1  = (const float*)d_in[6];
    const int N   = in_sizes[0] / MDIM;
    const int nnz = in_sizes[2];
    const size_t big = (size_t)N * MDIM;   // floats per node-major matrix

    float* Xt   = (float*)d_ws;            // [N,128] state (updated in place)
    float* Yt   = Xt + big;                // [N,128] W@X (node-major)
    float* Acc  = Yt + big;                // [N,128] spmm accumulator (Ut pre-loop)
    float* Bt   = Acc + big;               // [N,128] bias term
    float* Wpt  = Bt + big;                // [128,128] projected W, transposed
    float* Om1t = Wpt + 128 * 128;         // [128,128] Omega1 transposed
    unsigned* errMax = (unsigned*)(Om1t + 128 * 128);
    int* flag = (int*)(errMax + 1);

    init_kernel<<<1, 1, 0, stream>>>(flag, errMax);
    proj_w_kernel<<<128, 128, 0, stream>>>(W, Wpt);

    dim3 tb(32, 8);
    transpose_kernel<<<dim3((N + 31) / 32, 4), tb, 0, stream>>>(X0, Xt, MDIM, N);
    transpose_kernel<<<dim3((N + 31) / 32, 4), tb, 0, stream>>>(U, Acc, MDIM, N); // Ut
    transpose_kernel<<<dim3(4, 4), tb, 0, stream>>>(Om1, Om1t, MDIM, MDIM);

    const int gblocks = (N + 127) / 128;
    // S1t = Ut @ Om1^T  (into Yt), then Bt = A^T-scatter of S1t
    gemm_nodemajor_kernel<<<gblocks, 256, 0, stream>>>(Acc, Om1t, Yt, N, nullptr);
    zero_kernel<<<2048, 256, 0, stream>>>(Bt, big, nullptr, nullptr);
    spmm_kernel<<<4096, 256, 0, stream>>>(rows, cols, vals, Yt, Bt, nnz, nullptr);

    for (int it = 0; it < MAX_ITERS; ++it) {
        zero_kernel<<<2048, 256, 0, stream>>>(Acc, big, errMax, flag);
        gemm_nodemajor_kernel<<<gblocks, 256, 0, stream>>>(Xt, Wpt, Yt, N, flag);
        spmm_kernel<<<4096, 256, 0, stream>>>(rows, cols, vals, Yt, Acc, nnz, flag);
        update_kernel<<<2048, 256, 0, stream>>>(Xt, Acc, Bt, errMax, flag, big);
        check_kernel<<<1, 1, 0, stream>>>(errMax, flag);
    }
    // final transpose back to [128, N]
    transpose_kernel<<<dim3(4, (N + 31) / 32), tb, 0, stream>>>(Xt, (float*)d_out, N, MDIM);
}